// YOLOLoss_45354854645760
// MI455X (gfx1250) — compile-verified
//
#include <hip/hip_runtime.h>
#include <math.h>

// ---- problem constants (fixed by setup_inputs in the reference) ----
constexpr int B  = 64;
constexpr int H  = 80;
constexpr int W  = 80;
constexpr int HW = H * W;      // 6400
constexpr int C  = 85;         // 5 + 80 classes
constexpr int T  = 50;
constexpr int NOBJ   = B * T;        // 3200 per-object losses
constexpr int NVALS  = NOBJ + B;     // + 64 per-image noobj terms = 3264
constexpr int NTILES = NVALS / 64;   // 51 tiles of 64 for WMMA reduction
constexpr int SLICES = 8;            // noobj reduction slices per image
constexpr int SLICE_LEN = HW / SLICES; // 800

#define LAMBDA_COORD 5.0f
#define LAMBDA_NOOBJ 0.5f

// ---- workspace layout (bytes) ----
// [0, 13056)        : vals, float[3264]
// [13056, 15104)    : psum, float[B*SLICES]
// [15104, 17152)    : pcnt, float[B*SLICES]
// [20480, 430080)   : mask, uchar[B*HW]
constexpr size_t WS_VALS = 0;
constexpr size_t WS_PSUM = 13056;
constexpr size_t WS_PCNT = 15104;
constexpr size_t WS_MASK = 20480;

typedef __attribute__((ext_vector_type(2))) float v2f;
typedef __attribute__((ext_vector_type(8))) float v8f;

__device__ __forceinline__ float softplus_(float x) {
    // jax.nn.softplus(x) = max(x,0) + log1p(exp(-|x|))
    return log1pf(expf(-fabsf(x))) + fmaxf(x, 0.0f);
}

__global__ void zero_mask_kernel(unsigned int* __restrict__ mask_u32, int n_u32) {
    int i = blockIdx.x * blockDim.x + threadIdx.x;
    if (i < n_u32) mask_u32[i] = 0u;
}

// One thread per (b, t) object: mark cell + compute full per-object loss.
__global__ void obj_kernel(const float* __restrict__ pred,
                           const float* __restrict__ tbox,
                           const int*   __restrict__ tcls,
                           unsigned char* __restrict__ mask,
                           float* __restrict__ vals) {
    int idx = blockIdx.x * blockDim.x + threadIdx.x;
    if (idx >= NOBJ) return;
    int b = idx / T;

    const float* bx = tbox + (size_t)idx * 4;
    float cx = bx[0], cy = bx[1], w = bx[2], h = bx[3];
    float gxf = floorf(cx * (float)W);
    float gyf = floorf(cy * (float)H);
    int gi = (int)(gyf * (float)W + gxf);

    mask[(size_t)b * HW + gi] = 1;   // idempotent: races write same value

    const float* g = pred + ((size_t)b * HW + gi) * C;

    // coordinate (xy) loss: mean over 2 comps of (sigmoid - frac)^2
    float tx = cx * (float)W - gxf;
    float ty = cy * (float)H - gyf;
    float sx = 1.0f / (1.0f + expf(-g[0]));
    float sy = 1.0f / (1.0f + expf(-g[1]));
    float xy = 0.5f * ((sx - tx) * (sx - tx) + (sy - ty) * (sy - ty));

    // wh loss vs log targets
    float tw = logf(w * (float)W + 1e-16f);
    float th = logf(h * (float)H + 1e-16f);
    float wh = 0.5f * ((g[2] - tw) * (g[2] - tw) + (g[3] - th) * (g[3] - th));

    // objectness BCE with target 1: softplus(-p)
    float conf = softplus_(-g[4]);

    // class cross-entropy via stable logsumexp over the 80 class logits
    float m = -INFINITY;
    for (int j = 5; j < C; ++j) m = fmaxf(m, g[j]);
    float s = 0.0f;
    for (int j = 5; j < C; ++j) s += expf(g[j] - m);
    float lse = m + logf(s);
    float cls = lse - g[5 + tcls[idx]];

    vals[idx] = LAMBDA_COORD * (xy + wh) + conf + cls;
}

// One block per (image, slice): deterministic LDS tree reduction of
// softplus(conf) over no-object cells + the no-object count.
__global__ void noobj_partial_kernel(const float* __restrict__ pred,
                                     const unsigned char* __restrict__ mask,
                                     float* __restrict__ psum,
                                     float* __restrict__ pcnt) {
    __shared__ float ssum[256];
    __shared__ float scnt[256];
    int b     = blockIdx.x / SLICES;
    int slice = blockIdx.x % SLICES;
    int t     = threadIdx.x;
    int start = slice * SLICE_LEN;

    float s = 0.0f, c = 0.0f;
    for (int i = start + t; i < start + SLICE_LEN; i += 256) {
        if (!mask[(size_t)b * HW + i]) {
            s += softplus_(pred[((size_t)b * HW + i) * C + 4]);
            c += 1.0f;
        }
    }
    ssum[t] = s; scnt[t] = c;
    __syncthreads();
    for (int k = 128; k > 0; k >>= 1) {
        if (t < k) { ssum[t] += ssum[t + k]; scnt[t] += scnt[t + k]; }
        __syncthreads();
    }
    if (t == 0) { psum[blockIdx.x] = ssum[0]; pcnt[blockIdx.x] = scnt[0]; }
}

// Fold the per-slice partials into per-image noobj terms (fixed order).
__global__ void noobj_combine_kernel(const float* __restrict__ psum,
                                     const float* __restrict__ pcnt,
                                     float* __restrict__ vals) {
    int b = threadIdx.x;
    if (b >= B) return;
    float s = 0.0f, c = 0.0f;
    for (int k = 0; k < SLICES; ++k) {
        s += psum[b * SLICES + k];
        c += pcnt[b * SLICES + k];
    }
    vals[NOBJ + b] = LAMBDA_NOOBJ * s / fmaxf(c, 1.0f);
}

// Single-wave final reduction of 3264 values using the f32 matrix pipe:
// per tile of 64, A = 16x4 partials, B = ones(4x16), D[m][n] += rowsum_m.
// Grand total = (sum of D VGPRs in lane 0) + (same in lane 16).
__global__ void final_wmma_reduce_kernel(const float* __restrict__ vals,
                                         float* __restrict__ out) {
    int lane = threadIdx.x;             // 32 threads, EXEC all ones
    v8f acc = {};
    v2f ones; ones.x = 1.0f; ones.y = 1.0f;
    for (int tile = 0; tile < NTILES; ++tile) {
        v2f a;
        a.x = vals[tile * 64 + lane];
        a.y = vals[tile * 64 + 32 + lane];
        acc = __builtin_amdgcn_wmma_f32_16x16x4_f32(
            /*neg_a=*/false, a, /*neg_b=*/false, ones,
            /*c_mod=*/(short)0, acc, /*reuse_a=*/false, /*reuse_b=*/false);
    }
    float s = acc[0] + acc[1] + acc[2] + acc[3] + acc[4] + acc[5] + acc[6] + acc[7];
    float total = __shfl(s, 0) + __shfl(s, 16);
    if (lane == 0) out[0] = total / (float)NOBJ;
}

extern "C" void kernel_launch(void* const* d_in, const int* in_sizes, int n_in,
                              void* d_out, int out_size, void* d_ws, size_t ws_size,
                              hipStream_t stream) {
    (void)in_sizes; (void)n_in; (void)out_size; (void)ws_size;
    const float* pred = (const float*)d_in[0];
    const float* tbox = (const float*)d_in[1];
    const int*   tcls = (const int*)d_in[2];

    char* ws = (char*)d_ws;
    float*         vals = (float*)(ws + WS_VALS);
    float*         psum = (float*)(ws + WS_PSUM);
    float*         pcnt = (float*)(ws + WS_PCNT);
    unsigned char* mask = (unsigned char*)(ws + WS_MASK);
    float*         out  = (float*)d_out;

    const int mask_u32 = (B * HW) / 4;  // 102400
    zero_mask_kernel<<<(mask_u32 + 255) / 256, 256, 0, stream>>>(
        (unsigned int*)mask, mask_u32);

    obj_kernel<<<(NOBJ + 255) / 256, 256, 0, stream>>>(pred, tbox, tcls, mask, vals);

    noobj_partial_kernel<<<B * SLICES, 256, 0, stream>>>(pred, mask, psum, pcnt);

    noobj_combine_kernel<<<1, 64, 0, stream>>>(psum, pcnt, vals);

    final_wmma_reduce_kernel<<<1, 32, 0, stream>>>(vals, out);
}